// DeepGCNLayer_3951369912917
// MI455X (gfx1250) — compile-verified
//
#include <hip/hip_runtime.h>
#include <hip/hip_bf16.h>

typedef __attribute__((ext_vector_type(2))) float v2f;
typedef __attribute__((ext_vector_type(8))) float v8f;

#define NB 32
#define NPG 1024
#define NN (NB * NPG)          // 32768 nodes
#define KNN 16
#define CCH 128
#define NE (NN * KNN)          // 524288 edges
#define TP 132                 // LDS row pitch (floats), mult of 4, breaks bank conflicts

// ---------------------------------------------------------------- KNN
__global__ void k_knn(const float* __restrict__ pos, int* __restrict__ nbr) {
  __shared__ float sp[NPG * 3];
  const int g = blockIdx.x >> 2;              // 4 blocks of 256 per graph
  const int base = g * NPG;
  for (int i = threadIdx.x; i < NPG * 3; i += 256) sp[i] = pos[base * 3 + i];
  __syncthreads();
  const int q = ((blockIdx.x & 3) << 8) + threadIdx.x;   // local query id
  const float px = sp[q * 3 + 0], py = sp[q * 3 + 1], pz = sp[q * 3 + 2];
  const float sqi = px * px + py * py + pz * pz;
  float bd[KNN];
  int   bi[KNN];
#pragma unroll
  for (int t = 0; t < KNN; ++t) { bd[t] = 3.0e38f; bi[t] = 0; }
  for (int j = 0; j < NPG; ++j) {
    const float qx = sp[j * 3 + 0], qy = sp[j * 3 + 1], qz = sp[j * 3 + 2];
    const float sqj = qx * qx + qy * qy + qz * qz;
    const float d = sqi + sqj - 2.0f * (px * qx + py * qy + pz * qz);
    if (j == q) continue;                     // exclude self (loop=False)
    if (d < bd[KNN - 1]) {
      bd[KNN - 1] = d; bi[KNN - 1] = j;
#pragma unroll
      for (int t = KNN - 1; t > 0; --t) {
        if (bd[t] < bd[t - 1]) {
          float td = bd[t]; bd[t] = bd[t - 1]; bd[t - 1] = td;
          int   ti = bi[t]; bi[t] = bi[t - 1]; bi[t - 1] = ti;
        }
      }
    }
  }
  const int n = base + q;
#pragma unroll
  for (int t = 0; t < KNN; ++t) nbr[n * KNN + t] = base + bi[t];
}

// --------------------------------------------- Out(M x128) = X(M x128) @ W(128x128)
__global__ void k_gemm_xw(const float* __restrict__ X, const float* __restrict__ W,
                          float* __restrict__ Out) {
  __shared__ float sx[16 * TP];
  const int lane = threadIdx.x & 31;
  const int wv   = threadIdx.x >> 5;          // 8 waves -> 8 col blocks
  const int half = lane >> 4;
  const int ln   = lane & 15;
  const int col  = wv * 16 + ln;
  v2f bfr[32];
#pragma unroll
  for (int s = 0; s < 32; ++s) {              // B-frag: rows 4s+2h, 4s+2h+1
    const int r0 = 4 * s + 2 * half;
    bfr[s].x = W[r0 * CCH + col];
    bfr[s].y = W[(r0 + 1) * CCH + col];
  }
  const int rowbase = blockIdx.x * 16;
  for (int i = threadIdx.x; i < 16 * CCH / 4; i += 256) {
    const int k = (i * 4) >> 7, c = (i * 4) & 127;
    *(float4*)&sx[k * TP + c] = *(const float4*)&X[(rowbase + k) * CCH + c];
  }
  __syncthreads();
  v8f acc = {};
#pragma unroll
  for (int s = 0; s < 32; ++s) {
    v2f a = *(const v2f*)&sx[ln * TP + 4 * s + 2 * half];
    acc = __builtin_amdgcn_wmma_f32_16x16x4_f32(false, a, false, bfr[s],
                                                (short)0, acc, false, false);
  }
#pragma unroll
  for (int r = 0; r < 8; ++r)
    Out[(rowbase + r + 8 * half) * CCH + col] = acc[r];
}

// --------------------------------------------- BN1 stats over all edges (partials)
__global__ void k_stats_edges(const float* __restrict__ A, const float* __restrict__ Bm,
                              const float* __restrict__ b1, const int* __restrict__ nbr,
                              float* __restrict__ partial) {
  const int c = threadIdx.x;                  // 128 threads = channels
  const float bc = b1[c];
  float s = 0.f, s2 = 0.f;
  const int n0 = blockIdx.x * 64;             // 512 blocks * 64 nodes
  for (int n = n0; n < n0 + 64; ++n) {
    const float av = A[n * CCH + c] + bc;
    for (int k = 0; k < KNN; ++k) {
      const int j = nbr[n * KNN + k];
      const float v = av + Bm[j * CCH + c];
      s += v; s2 += v * v;
    }
  }
  partial[blockIdx.x * 256 + c]       = s;
  partial[blockIdx.x * 256 + 128 + c] = s2;
}

// --------------------------------------------- deterministic reduce + BN finalize
__global__ void k_reduce_stats(const float* __restrict__ partial, int G, float R,
                               const float* __restrict__ g, const float* __restrict__ be,
                               float* __restrict__ scale, float* __restrict__ shift) {
  const int c = threadIdx.x;                  // 128 threads
  float s = 0.f, s2 = 0.f;
  for (int b = 0; b < G; ++b) { s += partial[b * 256 + c]; s2 += partial[b * 256 + 128 + c]; }
  const float mean = s / R;
  const float var  = s2 / R - mean * mean;
  const float rstd = rsqrtf(var + 1e-5f);
  const float sc   = g[c] * rstd;
  scale[c] = sc;
  shift[c] = be[c] - mean * sc;
}

// --------------------------------------------- GEMM2 (H1 built on the fly in LDS)
// PASS 1: accumulate BN2 stats of Y2.  PASS 2: BN2+ReLU+max-over-K -> agg, BN3 stats.
template <int PASS>
__global__ void k_gemm2(const float* __restrict__ A, const float* __restrict__ Bm,
                        const float* __restrict__ b1, const float* __restrict__ scale1,
                        const float* __restrict__ shift1, const float* __restrict__ W2,
                        const float* __restrict__ b2, const float* __restrict__ scale2,
                        const float* __restrict__ shift2, const int* __restrict__ nbr,
                        float* __restrict__ agg, float* __restrict__ partial) {
  __shared__ float sh[16 * TP];
  __shared__ int   snbr[16];
  const int lane = threadIdx.x & 31;
  const int wv   = threadIdx.x >> 5;
  const int half = lane >> 4;
  const int ln   = lane & 15;
  const int col  = wv * 16 + ln;
  v2f bfr[32];
#pragma unroll
  for (int s = 0; s < 32; ++s) {
    const int r0 = 4 * s + 2 * half;
    bfr[s].x = W2[r0 * CCH + col];
    bfr[s].y = W2[(r0 + 1) * CCH + col];
  }
  const float b2c = b2[col];
  const float sc2 = (PASS == 2) ? scale2[col] : 0.f;
  const float sf2 = (PASS == 2) ? shift2[col] : 0.f;
  float s = 0.f, s2 = 0.f;
  for (int n = blockIdx.x; n < NN; n += gridDim.x) {
    if (threadIdx.x < 16) snbr[threadIdx.x] = nbr[n * KNN + threadIdx.x];
    __syncthreads();
    for (int i = threadIdx.x; i < 16 * CCH; i += 256) {   // H1 tile: 16 edges x 128 ch
      const int k = i >> 7, c = i & 127;
      const int j = snbr[k];
      const float v = A[n * CCH + c] + Bm[j * CCH + c] + b1[c];
      sh[k * TP + c] = fmaxf(v * scale1[c] + shift1[c], 0.f);
    }
    __syncthreads();
    v8f acc = {};
#pragma unroll
    for (int ss = 0; ss < 32; ++ss) {
      v2f a = *(const v2f*)&sh[ln * TP + 4 * ss + 2 * half];
      acc = __builtin_amdgcn_wmma_f32_16x16x4_f32(false, a, false, bfr[ss],
                                                  (short)0, acc, false, false);
    }
    if (PASS == 1) {
#pragma unroll
      for (int r = 0; r < 8; ++r) { const float y = acc[r] + b2c; s += y; s2 += y * y; }
    } else {
      float m = -3.0e38f;
#pragma unroll
      for (int r = 0; r < 8; ++r) {
        const float y = acc[r] + b2c;
        m = fmaxf(m, fmaxf(y * sc2 + sf2, 0.f));
      }
      m = fmaxf(m, __shfl_xor(m, 16, 32));                // combine row halves
      if (half == 0) { agg[n * CCH + col] = m; s += m; s2 += m * m; }
    }
    __syncthreads();
  }
  if (PASS == 1) { s += __shfl_xor(s, 16, 32); s2 += __shfl_xor(s2, 16, 32); }
  if (half == 0) {
    partial[blockIdx.x * 256 + col]       = s;
    partial[blockIdx.x * 256 + 128 + col] = s2;
  }
}

// --------------------------------------------- residual + BN3 + relu
__global__ void k_final(const float* __restrict__ agg, const float* __restrict__ x,
                        const float* __restrict__ scale3, const float* __restrict__ shift3,
                        float* __restrict__ out) {
  const int i = blockIdx.x * 256 + threadIdx.x;
  const int c = i & 127;
  out[i] = fmaxf(agg[i] * scale3[c] + shift3[c] + x[i], 0.f);
}

// ================================================================ host
extern "C" void kernel_launch(void* const* d_in, const int* in_sizes, int n_in,
                              void* d_out, int out_size, void* d_ws, size_t ws_size,
                              hipStream_t stream) {
  const float* x   = (const float*)d_in[0];
  const float* pos = (const float*)d_in[1];
  const float* W1  = (const float*)d_in[2];
  const float* b1  = (const float*)d_in[3];
  const float* g1  = (const float*)d_in[4];
  const float* be1 = (const float*)d_in[5];
  const float* W2  = (const float*)d_in[6];
  const float* b2  = (const float*)d_in[7];
  const float* g2  = (const float*)d_in[8];
  const float* be2 = (const float*)d_in[9];
  const float* gn  = (const float*)d_in[10];
  const float* bnb = (const float*)d_in[11];
  float* out = (float*)d_out;

  char* w = (char*)d_ws;
  int*   nbr   = (int*)(w);                                   // 2 MB
  float* A     = (float*)(w + (2u  << 20));                   // 16 MB
  float* Bm    = (float*)(w + (18u << 20));                   // 16 MB
  float* agg   = (float*)(w + (34u << 20));                   // 16 MB
  float* p1    = (float*)(w + (50u << 20));                   // 512 KB (512 blks)
  float* p2    = (float*)(w + (50u << 20) + (1u << 19));      // 1 MB   (1024 blks)
  float* p3    = (float*)(w + (50u << 20) + (3u << 19));      // 1 MB
  float* scl1  = (float*)(w + (50u << 20) + (5u << 19));
  float* shf1  = scl1 + 128;
  float* scl2  = scl1 + 256;
  float* shf2  = scl1 + 384;
  float* scl3  = scl1 + 512;
  float* shf3  = scl1 + 640;

  // 1) KNN
  k_knn<<<NN / 256, 256, 0, stream>>>(pos, nbr);
  // 2) A = X@W1_top ; Bm = X@W1_bot
  k_gemm_xw<<<NN / 16, 256, 0, stream>>>(x, W1, A);
  k_gemm_xw<<<NN / 16, 256, 0, stream>>>(x, W1 + CCH * CCH, Bm);
  // 3) BN1 stats + finalize
  k_stats_edges<<<512, 128, 0, stream>>>(A, Bm, b1, nbr, p1);
  k_reduce_stats<<<1, 128, 0, stream>>>(p1, 512, (float)NE, g1, be1, scl1, shf1);
  // 4) GEMM2 pass1 (BN2 stats) + finalize
  k_gemm2<1><<<1024, 256, 0, stream>>>(A, Bm, b1, scl1, shf1, W2, b2, scl2, shf2,
                                       nbr, agg, p2);
  k_reduce_stats<<<1, 128, 0, stream>>>(p2, 1024, (float)NE, g2, be2, scl2, shf2);
  // 5) GEMM2 pass2: BN2+ReLU+max-over-K -> agg, BN3 stats + finalize
  k_gemm2<2><<<1024, 256, 0, stream>>>(A, Bm, b1, scl1, shf1, W2, b2, scl2, shf2,
                                       nbr, agg, p3);
  k_reduce_stats<<<1, 128, 0, stream>>>(p3, 1024, (float)NN, gn, bnb, scl3, shf3);
  // 6) residual + BN3 + relu
  k_final<<<(NN * CCH) / 256, 256, 0, stream>>>(agg, x, scl3, shf3, out);
}